// GATGraphModel_57002805952823
// MI455X (gfx1250) — compile-verified
//
#include <hip/hip_runtime.h>
#include <hip/hip_bf16.h>

#define IN_CH 16
#define NH    12     // heads
#define DH    14     // dim per head
#define HD    168    // NH*DH
#define POOL  50
#define DNN   400
#define GF    45
#define KMAX  168    // max GEMM K (HD)
#define NT    4      // N-tiles per wave (64 output columns per block)

typedef float v2f __attribute__((ext_vector_type(2)));
typedef float v8f __attribute__((ext_vector_type(8)));

__device__ __forceinline__ float eluf(float x) { return x > 0.f ? x : (expf(x) - 1.f); }
__device__ __forceinline__ float lrelu(float x) { return x > 0.f ? x : 0.2f * x; }

__device__ __forceinline__ void atomicMaxF(float* addr, float val) {
    unsigned int* ua = (unsigned int*)addr;
    unsigned int old = *ua;
    while (__uint_as_float(old) < val) {
        unsigned int assumed = old;
        old = atomicCAS(ua, assumed, __float_as_uint(val));
        if (old == assumed) break;
    }
}

// ---------------------------------------------------------------------------
// Per-column batch-norm statistics: scale[c]=rsqrt(var+eps)*g[c],
// shift[c]=b[c]-mean*scale[c], cmean[c]=mean.  Element (i,c) at X[i*rs + c*cs].
// ---------------------------------------------------------------------------
__global__ void __launch_bounds__(256)
col_stats(const float* __restrict__ X, int n, long rs, long cs,
          const float* __restrict__ g, const float* __restrict__ b,
          float* __restrict__ scale, float* __restrict__ shift,
          float* __restrict__ cmean)
{
    int c = blockIdx.x;
    __shared__ float ssum[256], ssq[256];
    float s = 0.f, q = 0.f;
    for (int i = threadIdx.x; i < n; i += 256) {
        float v = X[(long)i * rs + (long)c * cs];
        s += v; q += v * v;
    }
    ssum[threadIdx.x] = s; ssq[threadIdx.x] = q;
    __syncthreads();
    for (int o = 128; o > 0; o >>= 1) {
        if (threadIdx.x < o) { ssum[threadIdx.x] += ssum[threadIdx.x + o]; ssq[threadIdx.x] += ssq[threadIdx.x + o]; }
        __syncthreads();
    }
    if (threadIdx.x == 0) {
        float mu  = ssum[0] / (float)n;
        float var = ssq[0] / (float)n - mu * mu;
        float rsd = rsqrtf(var + 1e-5f);
        float sc  = rsd * g[c];
        scale[c] = sc;
        shift[c] = b[c] - mu * sc;
        cmean[c] = mu;
    }
}

// ---------------------------------------------------------------------------
// OUT[M x 168] = BN(A[M x K]) @ W[K x 168] using V_WMMA_F32_16X16X4_F32.
// B tile is staged in LDS PRE-SWIZZLED into per-lane WMMA fragment pairs:
// one aligned ds_load_b64 per fragment, no register shuffling.
// CONTIG=true (mid layers, A row-major contiguous in k): A pair is one
// global_load_b64.  CONTIG=false (layer 0, A transposed with column stride
// ca): two scalar loads.  Each wave: 16 rows x 4 N-tiles (A fragment reused
// 4x per K-step).  8 waves/block, grid.y = 3.
// ---------------------------------------------------------------------------
template <bool CONTIG>
__global__ void __launch_bounds__(256)
gemm_bn_wmma(const float* __restrict__ A, long ra, long ca,
             const float* __restrict__ W, int K,
             const float* __restrict__ scale, const float* __restrict__ shift,
             float* __restrict__ OUT, int mtiles)
{
    __shared__ float Bs[(KMAX / 4) * NT * 32 * 2];   // per-lane (b.x,b.y) pairs
    __shared__ float Ss[KMAX];
    __shared__ float Ts[KMAX];

    const int tid = threadIdx.x;
    const int n0  = blockIdx.y * (16 * NT);
    const int KG  = K >> 2;                           // K groups of 4

    // cooperative staging: swizzled W fragments (zero-padded), BN scale/shift
    for (int idx = tid; idx < KG * NT * 32; idx += 256) {
        const int kg  = idx >> 7;                     // /(NT*32)
        const int rem = idx & 127;
        const int t   = rem >> 5;
        const int l   = rem & 31;
        const int k   = kg * 4 + ((l >> 4) << 1);     // k0 + khalf(lane)
        const int col = n0 + t * 16 + (l & 15);
        const bool ok = (col < HD);
        Bs[2 * idx]     = ok ? W[(long)k * HD + col]       : 0.f;
        Bs[2 * idx + 1] = ok ? W[(long)(k + 1) * HD + col] : 0.f;
    }
    for (int k = tid; k < K; k += 256) { Ss[k] = scale[k]; Ts[k] = shift[k]; }
    __syncthreads();

    const int wave  = tid >> 5;
    const int lane  = tid & 31;
    const int mtile = blockIdx.x * 8 + wave;
    if (mtile < mtiles) {                             // wave-uniform: EXEC all-1s
        const int m0    = mtile * 16;
        const int lmod  = lane & 15;
        const int khalf = (lane >> 4) * 2;
        const long mra  = (long)(m0 + lmod) * ra;
        const v2f* __restrict__ Bv = (const v2f*)Bs;

        v8f acc[NT];
#pragma unroll
        for (int t = 0; t < NT; ++t) acc[t] = (v8f){};

#pragma unroll 2
        for (int k0 = 0; k0 < K; k0 += 4) {
            const int ka = k0 + khalf;
            v2f a;
            if (CONTIG) {
                a = *(const v2f*)(A + mra + ka);       // ka even, row stride even
            } else {
                a.x = A[mra + (long)ka * ca];
                a.y = A[mra + (long)(ka + 1) * ca];
            }
            a.x = a.x * Ss[ka]     + Ts[ka];
            a.y = a.y * Ss[ka + 1] + Ts[ka + 1];
            const int base = (k0 >> 2) * (NT * 32) + lane;
#pragma unroll
            for (int t = 0; t < NT; ++t) {
                v2f b = Bv[base + t * 32];
                acc[t] = __builtin_amdgcn_wmma_f32_16x16x4_f32(false, a, false, b,
                                                               (short)0, acc[t], false, false);
            }
        }

#pragma unroll
        for (int t = 0; t < NT; ++t) {
            const int ncol = n0 + t * 16 + lmod;
            if (ncol < HD) {
#pragma unroll
                for (int r = 0; r < 8; ++r) {
                    const int row = m0 + r + (lane >> 4) * 8;
                    OUT[(long)row * HD + ncol] = acc[t][r];
                }
            }
        }
    }
}

// ---------------------------------------------------------------------------
// Per-node attention scores: a_s[n,h] = sum_d h[n,h,d]*asrc[h,d] (same for dst)
// ---------------------------------------------------------------------------
__global__ void __launch_bounds__(256)
attn_scores(const float* __restrict__ HT,
            const float* __restrict__ asrc, const float* __restrict__ adst,
            float* __restrict__ AS, float* __restrict__ AD, int n)
{
    int tid = blockIdx.x * 256 + threadIdx.x;
    if (tid >= n * NH) return;
    int node = tid / NH, h = tid % NH;
    const float* hp = HT + (long)node * HD + h * DH;
    float s = 0.f, d = 0.f;
#pragma unroll
    for (int k = 0; k < DH; ++k) { s += hp[k] * asrc[h * DH + k]; d += hp[k] * adst[h * DH + k]; }
    AS[tid] = s; AD[tid] = d;
}

__global__ void __launch_bounds__(256)
init_layer(float* __restrict__ M, float* __restrict__ DEN, float* __restrict__ AGG, int n)
{
    long tid = (long)blockIdx.x * 256 + threadIdx.x;
    if (tid < (long)n * NH) { M[tid] = -1e30f; DEN[tid] = 0.f; }
    if (tid < (long)n * HD) AGG[tid] = 0.f;
}

// pass 1: segment max of leaky(a_s[src]+a_d[dst]) over dst
__global__ void __launch_bounds__(256)
edge_max(const int* __restrict__ ei, int E, int n,
         const float* __restrict__ AS, const float* __restrict__ AD,
         float* __restrict__ M)
{
    long e = (long)blockIdx.x * 256 + threadIdx.x;
    long EP = (long)E + n;
    if (e >= EP) return;
    int src = (e < E) ? ei[e]     : (int)(e - E);
    int dst = (e < E) ? ei[E + e] : (int)(e - E);
#pragma unroll
    for (int h = 0; h < NH; ++h) {
        float v = lrelu(AS[src * NH + h] + AD[dst * NH + h]);
        atomicMaxF(&M[dst * NH + h], v);
    }
}

// pass 2: exponentials + segment denominator
__global__ void __launch_bounds__(256)
edge_expsum(const int* __restrict__ ei, int E, int n,
            const float* __restrict__ AS, const float* __restrict__ AD,
            const float* __restrict__ M,
            float* __restrict__ EEXP, float* __restrict__ DEN)
{
    long e = (long)blockIdx.x * 256 + threadIdx.x;
    long EP = (long)E + n;
    if (e >= EP) return;
    int src = (e < E) ? ei[e]     : (int)(e - E);
    int dst = (e < E) ? ei[E + e] : (int)(e - E);
#pragma unroll
    for (int h = 0; h < NH; ++h) {
        float v  = lrelu(AS[src * NH + h] + AD[dst * NH + h]);
        float ex = expf(v - M[dst * NH + h]);
        EEXP[e * NH + h] = ex;
        atomicAdd(&DEN[dst * NH + h], ex);
    }
}

// pass 3: agg[dst,c] += h[src,c] * alpha.  One edge per 192-thread block:
// scalar src/dst loads, lane-coalesced gather + atomic scatter, no 64-bit div.
__global__ void __launch_bounds__(192)
edge_aggregate(const int* __restrict__ ei, int E, int n,
               const float* __restrict__ HT,
               const float* __restrict__ EEXP, const float* __restrict__ DEN,
               float* __restrict__ AGG)
{
    const long e = blockIdx.x;
    const int c = threadIdx.x;
    if (c >= HD) return;
    const int src = (e < E) ? ei[e]     : (int)(e - E);
    const int dst = (e < E) ? ei[E + e] : (int)(e - E);
    const int h = c / DH;
    const float alpha = EEXP[e * NH + h] / (DEN[(long)dst * NH + h] + 1e-16f);
    atomicAdd(&AGG[(long)dst * HD + c], HT[(long)src * HD + c] * alpha);
}

// concat layers: out = elu(agg + bias)
__global__ void __launch_bounds__(256)
bias_elu(const float* __restrict__ AGG, const float* __restrict__ bias,
         float* __restrict__ OUT, int n)
{
    long tid = (long)blockIdx.x * 256 + threadIdx.x;
    if (tid >= (long)n * HD) return;
    int c = (int)(tid % HD);
    OUT[tid] = eluf(AGG[tid] + bias[c]);
}

// last layer: mean over heads + bias (no elu)
__global__ void __launch_bounds__(256)
finalize_last(const float* __restrict__ AGG, const float* __restrict__ bias,
              float* __restrict__ OUT14, int n)
{
    long tid = (long)blockIdx.x * 256 + threadIdx.x;
    if (tid >= (long)n * DH) return;
    int node = (int)(tid / DH), d = (int)(tid % DH);
    float s = 0.f;
#pragma unroll
    for (int h = 0; h < NH; ++h) s += AGG[(long)node * HD + h * DH + d];
    OUT14[tid] = s * (1.f / (float)NH) + bias[d];
}

// ---------------------------------------------------------------------------
// Head: BN-pool mean -> ugp1 -> mean-pool(=linear) -> ugp2 -> concat gf1 -> DNN
// Single block; exploits mean(BN(h)@W+b) == mean(BN(h))@W+b.
// ---------------------------------------------------------------------------
__global__ void __launch_bounds__(512)
head_mlp(const float* __restrict__ cmean, const float* __restrict__ scale,
         const float* __restrict__ shift,
         const float* __restrict__ u1W, const float* __restrict__ u1b,
         const float* __restrict__ u2W, const float* __restrict__ u2b,
         const float* __restrict__ gf1,
         const float* __restrict__ d0W, const float* __restrict__ d0b,
         const float* __restrict__ dmW, const float* __restrict__ dmb,
         const float* __restrict__ dlW, const float* __restrict__ dlb,
         float* __restrict__ out)
{
    __shared__ float hbar[DH];
    __shared__ float p1[POOL];
    __shared__ float z[POOL + GF];
    __shared__ float t0[DNN], t1[DNN];
    __shared__ float red[512];
    int tid = threadIdx.x;
    if (tid < DH) hbar[tid] = cmean[tid] * scale[tid] + shift[tid];
    __syncthreads();
    if (tid < POOL) {
        float s = u1b[tid];
        for (int d = 0; d < DH; ++d) s += hbar[d] * u1W[d * POOL + tid];
        p1[tid] = s;
    }
    __syncthreads();
    if (tid < POOL) {
        float s = u2b[tid];
        for (int k = 0; k < POOL; ++k) s += p1[k] * u2W[k * POOL + tid];
        z[tid] = s;
    }
    if (tid >= POOL && tid < POOL + GF) z[tid] = gf1[tid - POOL];
    __syncthreads();
    if (tid < DNN) {
        float s = d0b[tid];
        for (int i = 0; i < POOL + GF; ++i) s += z[i] * d0W[i * DNN + tid];
        t0[tid] = eluf(s);
    }
    __syncthreads();
    float* cur = t0; float* nxt = t1;
    for (int L = 0; L < 6; ++L) {
        if (tid < DNN) {
            float s = dmb[L * DNN + tid];
            const float* Wl = dmW + (size_t)L * DNN * DNN;
            for (int i = 0; i < DNN; ++i) s += cur[i] * Wl[i * DNN + tid];
            nxt[tid] = eluf(eluf(s));
        }
        __syncthreads();
        float* tmp = cur; cur = nxt; nxt = tmp;
    }
    red[tid] = (tid < DNN) ? cur[tid] * dlW[tid] : 0.f;
    __syncthreads();
    for (int o = 256; o > 0; o >>= 1) {
        if (tid < o) red[tid] += red[tid + o];
        __syncthreads();
    }
    if (tid == 0) {
        float s = red[0] + dlb[0];
        out[0] = 1.f / (1.f + expf(-s));
    }
}

// ---------------------------------------------------------------------------
extern "C" void kernel_launch(void* const* d_in, const int* in_sizes, int n_in,
                              void* d_out, int out_size, void* d_ws, size_t ws_size,
                              hipStream_t stream)
{
    const float* x        = (const float*)d_in[0];   // [16, N]
    const float* gf1      = (const float*)d_in[2];
    const float* bn_in_g  = (const float*)d_in[3];
    const float* bn_in_b  = (const float*)d_in[4];
    const float* gat0_W   = (const float*)d_in[5];
    const float* gat0_as  = (const float*)d_in[6];
    const float* gat0_ad  = (const float*)d_in[7];
    const float* gat0_b   = (const float*)d_in[8];
    const float* gm_W     = (const float*)d_in[9];
    const float* gm_as    = (const float*)d_in[10];
    const float* gm_ad    = (const float*)d_in[11];
    const float* gm_b     = (const float*)d_in[12];
    const float* bnm_g    = (const float*)d_in[13];
    const float* bnm_b    = (const float*)d_in[14];
    const float* gl_W     = (const float*)d_in[15];
    const float* gl_as    = (const float*)d_in[16];
    const float* gl_ad    = (const float*)d_in[17];
    const float* gl_b     = (const float*)d_in[18];
    const float* bnp_g    = (const float*)d_in[19];
    const float* bnp_b    = (const float*)d_in[20];
    const float* u1W      = (const float*)d_in[21];
    const float* u1b      = (const float*)d_in[22];
    const float* u2W      = (const float*)d_in[23];
    const float* u2b      = (const float*)d_in[24];
    const float* d0W      = (const float*)d_in[25];
    const float* d0b      = (const float*)d_in[26];
    const float* dmW      = (const float*)d_in[27];
    const float* dmb      = (const float*)d_in[28];
    const float* dlW      = (const float*)d_in[29];
    const float* dlb      = (const float*)d_in[30];
    const int*   ei       = (const int*)d_in[31];    // [2, E]

    const int N = in_sizes[0] / IN_CH;
    const int E = in_sizes[31] / 2;
    const long EP = (long)E + N;

    float* WS = (float*)d_ws;
    size_t off = 0;
    float* ABUF  = WS + off; off += (size_t)N * HD;
    float* HT    = WS + off; off += (size_t)N * HD;
    float* AGG   = WS + off; off += (size_t)N * HD;
    float* EEXP  = WS + off; off += (size_t)EP * NH;
    float* AS    = WS + off; off += (size_t)N * NH;
    float* AD    = WS + off; off += (size_t)N * NH;
    float* MB    = WS + off; off += (size_t)N * NH;
    float* DEN   = WS + off; off += (size_t)N * NH;
    float* SCALE = WS + off; off += 256;
    float* SHIFT = WS + off; off += 256;
    float* CMEAN = WS + off; off += 256;
    float* OUT14 = WS + off; off += (size_t)N * DH;

    const int mtiles = (N + 15) / 16;
    dim3 ggrid((mtiles + 7) / 8, (HD + 16 * NT - 1) / (16 * NT));   // (782, 3)
    const int nbNH  = (int)(((long)N * NH + 255) / 256);
    const int nbHD  = (int)(((long)N * HD + 255) / 256);
    const int nbDH  = (int)(((long)N * DH + 255) / 256);
    const int nbE   = (int)((EP + 255) / 256);

    for (int layer = 0; layer < 8; ++layer) {
        const float *W, *as_, *ad_, *bg, *bb;
        const float* A; long ra, ca; int K;
        if (layer == 0) {
            W = gat0_W; as_ = gat0_as; ad_ = gat0_ad;
            bg = bn_in_g; bb = bn_in_b;
            A = x; ra = 1; ca = N; K = IN_CH;
        } else if (layer <= 6) {
            int i = layer - 1;
            W  = gm_W + (size_t)i * HD * HD;
            as_ = gm_as + (size_t)i * NH * DH;
            ad_ = gm_ad + (size_t)i * NH * DH;
            bg = bnm_g + (size_t)i * HD; bb = bnm_b + (size_t)i * HD;
            A = ABUF; ra = HD; ca = 1; K = HD;
        } else {
            W = gl_W; as_ = gl_as; ad_ = gl_ad;
            bg = bnm_g + (size_t)6 * HD; bb = bnm_b + (size_t)6 * HD;
            A = ABUF; ra = HD; ca = 1; K = HD;
        }

        // BatchNorm statistics over the K input columns
        col_stats<<<K, 256, 0, stream>>>(A, N, ra, ca, bg, bb, SCALE, SHIFT, CMEAN);
        // Fused BN + GEMM (WMMA f32 16x16x4, swizzled LDS B fragments)
        if (layer == 0)
            gemm_bn_wmma<false><<<ggrid, 256, 0, stream>>>(A, ra, ca, W, K, SCALE, SHIFT, HT, mtiles);
        else
            gemm_bn_wmma<true><<<ggrid, 256, 0, stream>>>(A, ra, ca, W, K, SCALE, SHIFT, HT, mtiles);
        // Attention scores
        attn_scores<<<nbNH, 256, 0, stream>>>(HT, as_, ad_, AS, AD, N);
        // Segment softmax + aggregation over edges (incl. implicit self-loops)
        init_layer<<<nbHD, 256, 0, stream>>>(MB, DEN, AGG, N);
        edge_max<<<nbE, 256, 0, stream>>>(ei, E, N, AS, AD, MB);
        edge_expsum<<<nbE, 256, 0, stream>>>(ei, E, N, AS, AD, MB, EEXP, DEN);
        edge_aggregate<<<(int)EP, 192, 0, stream>>>(ei, E, N, HT, EEXP, DEN, AGG);

        if (layer < 7) {
            const float* bias = (layer == 0) ? gat0_b : gm_b + (size_t)(layer - 1) * HD;
            bias_elu<<<nbHD, 256, 0, stream>>>(AGG, bias, ABUF, N);
        } else {
            finalize_last<<<nbDH, 256, 0, stream>>>(AGG, gl_b, OUT14, N);
        }
    }

    // Pooling BN stats on [N,14], then the whole MLP head in one block
    col_stats<<<DH, 256, 0, stream>>>(OUT14, N, DH, 1, bnp_g, bnp_b, SCALE, SHIFT, CMEAN);
    head_mlp<<<1, 512, 0, stream>>>(CMEAN, SCALE, SHIFT, u1W, u1b, u2W, u2b, gf1,
                                    d0W, d0b, dmW, dmb, dlW, dlb, (float*)d_out);
}